// SelfAttention_14439680049580
// MI455X (gfx1250) — compile-verified
//
#include <hip/hip_runtime.h>
#include <math.h>

typedef __attribute__((ext_vector_type(2))) float v2f;
typedef __attribute__((ext_vector_type(8))) float v8f;

#define NB 4
#define NL 512
#define ND 128

// Hardware tanh: CDNA5 (gfx1250) has V_TANH_F32 on the transcendental pipe,
// which co-executes with the main VALU. Avoid the ocml software tanh.
__device__ __forceinline__ float htanh(float x) {
#if __has_builtin(__builtin_amdgcn_tanhf)
  return __builtin_amdgcn_tanhf(x);
#else
  float r;
  asm("v_tanh_f32 %0, %1" : "=v"(r) : "v"(x));
  return r;
#endif
}

// ---------------------------------------------------------------------------
// Kernel 1: AB[m, 0:256] = [ h@W1^T | h@W2^T ], m in [0,2048)
// GEMM M=2048, N=256, K=128 with V_WMMA_F32_16X16X4_F32.
// W1[e,d] = W[e*256 + d], W2[e,d] = W[e*256 + 128 + d]  (both K-major per row).
// ---------------------------------------------------------------------------
__global__ void proj_wmma(const float* __restrict__ h,
                          const float* __restrict__ W,
                          float* __restrict__ AB) {
  const int wave  = (blockIdx.x * blockDim.x + threadIdx.x) >> 5; // 0..2047
  const int lane  = threadIdx.x & 31;
  const int mtile = wave >> 4;   // 0..127
  const int ntile = wave & 15;   // 0..15
  const int mn    = lane & 15;
  const int hi    = lane >> 4;   // 0/1 -> K offset 0/2 in fragment

  const float* hrow = h + (size_t)(mtile * 16 + mn) * ND;        // A row (K-major)
  const int    n    = ntile * 16 + mn;
  const float* wrow = W + (size_t)(n & 127) * 256 + (n >> 7) * 128; // B col source

  v8f c = {};
  for (int kk = 0; kk < ND; kk += 4) {
    v2f a, b;
    a.x = hrow[kk + 2 * hi];
    a.y = hrow[kk + 2 * hi + 1];
    b.x = wrow[kk + 2 * hi];
    b.y = wrow[kk + 2 * hi + 1];
    c = __builtin_amdgcn_wmma_f32_16x16x4_f32(false, a, false, b,
                                              (short)0, c, false, false);
  }
  // C layout: VGPR r -> row hi*8 + r, col lane&15
  float* o = AB + (size_t)(mtile * 16 + hi * 8) * 256 + ntile * 16 + mn;
#pragma unroll
  for (int r = 0; r < 8; ++r) o[(size_t)r * 256] = c[r];
}

// ---------------------------------------------------------------------------
// Kernel 2: S[b,i,j] = sum_d v[d] * tanh(a[b,i,d] + bproj[b,j,d])
// 32x32 (i,j) tile per block, a/b tiles staged in padded LDS.
// Inner loop per element: v_add_f32 + v_tanh_f32 (trans) + v_fmac_f32 (VALU),
// letting the trans and VALU pipes co-execute.
// ---------------------------------------------------------------------------
__global__ void scores_tanh(const float* __restrict__ AB,
                            const float* __restrict__ v,
                            float* __restrict__ S) {
  __shared__ float as[32][129];   // pad to avoid 32-way bank conflicts
  __shared__ float bs[32][129];
  __shared__ float vs[128];

  const int blk = blockIdx.x;       // 0..1023
  const int b   = blk >> 8;         // batch
  const int it  = (blk >> 4) & 15;  // i tile (32 rows)
  const int jt  = blk & 15;         // j tile (32 cols)
  const int t   = threadIdx.x;      // 0..255

  for (int idx = t; idx < 32 * 128; idx += 256) {
    const int r = idx >> 7, d = idx & 127;
    as[r][d] = AB[((size_t)(b * NL + it * 32 + r)) * 256 + d];
    bs[r][d] = AB[((size_t)(b * NL + jt * 32 + r)) * 256 + 128 + d];
  }
  if (t < 128) vs[t] = v[t];
  __syncthreads();

  const int il = t & 31;   // lane == local i (conflict-free via pad)
  const int jg = t >> 5;   // wave id == j group (broadcast b reads)
  float acc0 = 0.f, acc1 = 0.f, acc2 = 0.f, acc3 = 0.f;
  for (int d = 0; d < 128; ++d) {
    const float av = as[il][d];
    const float vv = vs[d];
    acc0 += vv * htanh(av + bs[jg * 4 + 0][d]);
    acc1 += vv * htanh(av + bs[jg * 4 + 1][d]);
    acc2 += vv * htanh(av + bs[jg * 4 + 2][d]);
    acc3 += vv * htanh(av + bs[jg * 4 + 3][d]);
  }
  float* srow = S + ((size_t)b * NL + it * 32 + il) * NL + jt * 32 + jg * 4;
  srow[0] = acc0; srow[1] = acc1; srow[2] = acc2; srow[3] = acc3;
}

// ---------------------------------------------------------------------------
// Kernel 3: softmax over axis=1 (i) -> column-wise, in place.
// One wave32 per (b,j) column; 16 elements per lane; shfl reductions.
// ---------------------------------------------------------------------------
__global__ void softmax_cols(float* __restrict__ S) {
  const int wave = (blockIdx.x * blockDim.x + threadIdx.x) >> 5; // 0..2047
  const int lane = threadIdx.x & 31;
  const int b    = wave >> 9;
  const int j    = wave & (NL - 1);
  float* col = S + (size_t)b * NL * NL + j;

  float vals[16];
  float m = -INFINITY;
#pragma unroll
  for (int k = 0; k < 16; ++k) {
    vals[k] = col[(size_t)(lane + 32 * k) * NL];
    m = fmaxf(m, vals[k]);
  }
#pragma unroll
  for (int off = 16; off > 0; off >>= 1)
    m = fmaxf(m, __shfl_xor(m, off, 32));

  float s = 0.f;
#pragma unroll
  for (int k = 0; k < 16; ++k) {
    vals[k] = __expf(vals[k] - m);
    s += vals[k];
  }
#pragma unroll
  for (int off = 16; off > 0; off >>= 1)
    s += __shfl_xor(s, off, 32);

  const float inv = 1.0f / s;
#pragma unroll
  for (int k = 0; k < 16; ++k)
    col[(size_t)(lane + 32 * k) * NL] = vals[k] * inv;
}

// ---------------------------------------------------------------------------
// Kernel 4: out[b,i,d] = sum_j attn[b,i,j] * h[b,j,d]
// Per batch GEMM M=512, N=128, K=512 with f32 WMMA; h[b] is already K-major.
// ---------------------------------------------------------------------------
__global__ void out_wmma(const float* __restrict__ S,
                         const float* __restrict__ h,
                         float* __restrict__ out) {
  const int wave  = (blockIdx.x * blockDim.x + threadIdx.x) >> 5; // 0..1023
  const int lane  = threadIdx.x & 31;
  const int b     = wave >> 8;        // 256 tiles per batch
  const int tile  = wave & 255;
  const int mtile = tile >> 3;        // 0..31
  const int ntile = tile & 7;         // 0..7
  const int mn    = lane & 15;
  const int hi    = lane >> 4;

  const float* arow = S + ((size_t)b * NL + mtile * 16 + mn) * NL;
  const float* bcol = h + (size_t)b * NL * ND + ntile * 16 + mn;

  v8f c = {};
  for (int kk = 0; kk < NL; kk += 4) {
    v2f a, bb;
    a.x  = arow[kk + 2 * hi];
    a.y  = arow[kk + 2 * hi + 1];
    bb.x = bcol[(size_t)(kk + 2 * hi) * ND];
    bb.y = bcol[(size_t)(kk + 2 * hi + 1) * ND];
    c = __builtin_amdgcn_wmma_f32_16x16x4_f32(false, a, false, bb,
                                              (short)0, c, false, false);
  }
  float* o = out + (size_t)b * NL * ND + (size_t)(mtile * 16 + hi * 8) * ND +
             ntile * 16 + mn;
#pragma unroll
  for (int r = 0; r < 8; ++r) o[(size_t)r * ND] = c[r];
}

// ---------------------------------------------------------------------------
extern "C" void kernel_launch(void* const* d_in, const int* in_sizes, int n_in,
                              void* d_out, int out_size, void* d_ws, size_t ws_size,
                              hipStream_t stream) {
  const float* h = (const float*)d_in[0];   // (4,512,128)
  const float* W = (const float*)d_in[1];   // (128,256)
  const float* v = (const float*)d_in[2];   // (128,)
  float* out = (float*)d_out;               // (4,512,128)

  float* AB = (float*)d_ws;                 // 2048*256 floats = 2 MB
  float* S  = AB + (size_t)2048 * 256;      // 4*512*512 floats = 4 MB

  proj_wmma   <<<512,  128, 0, stream>>>(h, W, AB);
  scores_tanh <<<1024, 256, 0, stream>>>(AB, v, S);
  softmax_cols<<<256,  256, 0, stream>>>(S);
  out_wmma    <<<256,  128, 0, stream>>>(S, h, out);
}